// OUR_88699664597868
// MI455X (gfx1250) — compile-verified
//
#include <hip/hip_runtime.h>
#include <math.h>

#define BATCH 8
#define N_IN_PC 2048
#define N_COARSE 1024
#define N_FINE 16384
#define IMGSZ 224
#define FPS_NS 2048

typedef float v2f __attribute__((ext_vector_type(2)));
typedef float v8f __attribute__((ext_vector_type(8)));

// ---------------------------------------------------------------------------
// Farthest point sampling: one 1024-thread block per batch, 16 pts/thread.
// Exact jnp.argmax semantics (first index wins on ties).
// ---------------------------------------------------------------------------
__global__ __launch_bounds__(1024) void fps_kernel(const float* __restrict__ pf,
                                                   float* __restrict__ samp) {
    const int b = blockIdx.x;
    const int t = threadIdx.x;
    const float* P = pf + (size_t)b * N_FINE * 3;
    float* S = samp + (size_t)b * FPS_NS * 3;

    float px[16], py[16], pz[16], dm[16];
#pragma unroll
    for (int k = 0; k < 16; ++k) {
        const int i = t * 16 + k;
        px[k] = P[i * 3 + 0];
        py[k] = P[i * 3 + 1];
        pz[k] = P[i * 3 + 2];
        dm[k] = 3.4028234663852886e38f;
    }

    __shared__ float lx, ly, lz;
    __shared__ float wbd[32];
    __shared__ int   wbi[32];
    __shared__ int   sBest;

    if (t == 0) {
        lx = P[0]; ly = P[1]; lz = P[2];
        S[0] = lx; S[1] = ly; S[2] = lz;
    }
    __syncthreads();

    const int lane = t & 31;
    const int wv   = t >> 5;

    for (int it = 1; it < FPS_NS; ++it) {
        const float cx = lx, cy = ly, cz = lz;
        float bd = -1.0f;
        int   bi = 0x7fffffff;
#pragma unroll
        for (int k = 0; k < 16; ++k) {
            const float dx = px[k] - cx, dy = py[k] - cy, dz = pz[k] - cz;
            float d = dx * dx + dy * dy + dz * dz;
            d = fminf(dm[k], d);
            dm[k] = d;
            if (d > bd) { bd = d; bi = t * 16 + k; }   // strict > keeps lowest k
        }
        // wave32 argmax (keep lowest index on ties)
        for (int off = 16; off > 0; off >>= 1) {
            const float od = __shfl_down(bd, off, 32);
            const int   oi = __shfl_down(bi, off, 32);
            if (od > bd || (od == bd && oi < bi)) { bd = od; bi = oi; }
        }
        if (lane == 0) { wbd[wv] = bd; wbi[wv] = bi; }
        __syncthreads();
        if (wv == 0) {
            bd = wbd[lane];
            bi = wbi[lane];
            for (int off = 16; off > 0; off >>= 1) {
                const float od = __shfl_down(bd, off, 32);
                const int   oi = __shfl_down(bi, off, 32);
                if (od > bd || (od == bd && oi < bi)) { bd = od; bi = oi; }
            }
            if (lane == 0) sBest = bi;
        }
        __syncthreads();
        const int bestIdx = sBest;
        if ((bestIdx >> 4) == t) {
            const int k = bestIdx & 15;
            lx = px[k]; ly = py[k]; lz = pz[k];
            S[it * 3 + 0] = px[k];
            S[it * 3 + 1] = py[k];
            S[it * 3 + 2] = pz[k];
        }
        __syncthreads();
    }
}

// ---------------------------------------------------------------------------
// Density loss: per batch, 2048x2048 knn(16), mean of ranks 1..15 per point,
// emit per-batch sum / sumsq of those means (deterministic tree reduce).
// ---------------------------------------------------------------------------
__global__ __launch_bounds__(1024) void density_kernel(const float* __restrict__ samp,
                                                       float* __restrict__ dsum,
                                                       float* __restrict__ dssq) {
    const int b = blockIdx.x;
    const int t = threadIdx.x;
    __shared__ float sx[FPS_NS], sy[FPS_NS], sz[FPS_NS];
    const float* S = samp + (size_t)b * FPS_NS * 3;
    for (int i = t; i < FPS_NS; i += 1024) {
        sx[i] = S[i * 3 + 0];
        sy[i] = S[i * 3 + 1];
        sz[i] = S[i * 3 + 2];
    }
    __syncthreads();

    float lsum = 0.0f, lssq = 0.0f;
    for (int p = t; p < FPS_NS; p += 1024) {
        const float qx = sx[p], qy = sy[p], qz = sz[p];
        float a[16];
#pragma unroll
        for (int k = 0; k < 16; ++k) a[k] = 3.4028234663852886e38f;
        for (int j = 0; j < FPS_NS; ++j) {
            const float dx = sx[j] - qx, dy = sy[j] - qy, dz = sz[j] - qz;
            const float d = sqrtf(dx * dx + dy * dy + dz * dz);
            if (d < a[15]) {               // sorted insert, fully register-resident
                float cur = d;
#pragma unroll
                for (int k = 0; k < 16; ++k) {
                    const float mn = fminf(cur, a[k]);
                    const float mx = fmaxf(cur, a[k]);
                    a[k] = mn;
                    cur  = mx;
                }
            }
        }
        float s = 0.0f;
#pragma unroll
        for (int k = 1; k < 16; ++k) s += a[k];   // drop self (rank 0)
        const float m = s * (1.0f / 15.0f);
        lsum += m;
        lssq += m * m;
    }
    __syncthreads();
    sx[t] = lsum;
    sy[t] = lssq;
    __syncthreads();
    for (int off = 512; off > 0; off >>= 1) {
        if (t < off) { sx[t] += sx[t + off]; sy[t] += sy[t + off]; }
        __syncthreads();
    }
    if (t == 0) { dsum[b] = sx[0]; dssq[b] = sy[0]; }
}

// ---------------------------------------------------------------------------
// dense std: per-batch sum / sumsq over 16384 values
// ---------------------------------------------------------------------------
__global__ __launch_bounds__(1024) void dense_kernel(const float* __restrict__ dense,
                                                     float* __restrict__ dsum,
                                                     float* __restrict__ dssq) {
    const int b = blockIdx.x;
    const int t = threadIdx.x;
    const float* D = dense + (size_t)b * N_FINE;
    float s = 0.0f, q = 0.0f;
    for (int i = t; i < N_FINE; i += 1024) {
        const float v = D[i];
        s += v;
        q += v * v;
    }
    __shared__ float r1[1024], r2[1024];
    r1[t] = s;
    r2[t] = q;
    __syncthreads();
    for (int off = 512; off > 0; off >>= 1) {
        if (t < off) { r1[t] += r1[t + off]; r2[t] += r2[t + off]; }
        __syncthreads();
    }
    if (t == 0) { dsum[b] = r1[0]; dssq[b] = r2[0]; }
}

// ---------------------------------------------------------------------------
// Pack targets once: Tp[i] = (-2x, -2y, -2z, |p|^2)  (B-operand, loop-invariant)
// ---------------------------------------------------------------------------
__global__ __launch_bounds__(256) void pack_kernel(const float* __restrict__ T,
                                                   int n,
                                                   float* __restrict__ Tp) {
    const int i = blockIdx.x * blockDim.x + threadIdx.x;
    if (i < n) {
        const float x = T[i * 3 + 0];
        const float y = T[i * 3 + 1];
        const float z = T[i * 3 + 2];
        Tp[i * 4 + 0] = -2.0f * x;
        Tp[i * 4 + 1] = -2.0f * y;
        Tp[i * 4 + 2] = -2.0f * z;
        Tp[i * 4 + 3] = x * x + y * y + z * z;
    }
}

// ---------------------------------------------------------------------------
// Partial chamfer via V_WMMA_F32_16X16X4_F32.
// One wave handles 16 query rows; D tile = |b|^2 - 2 a.b ; add |a|^2 after min.
// A layout: lane L -> M=L&15, K=v+2*(L>>4). B layout: N=L&15, K=v+2*(L>>4).
// D layout: VGPR v, lanes 0-15 -> M=v, lanes 16-31 -> M=v+8.
// B operand comes pre-packed: lane loads exactly its two K-components (b64).
// ---------------------------------------------------------------------------
__global__ __launch_bounds__(256) void chamfer_kernel(const float* __restrict__ Q,
                                                      const float* __restrict__ Tpack,
                                                      int nt,
                                                      float* __restrict__ part) {
    const int wid  = (blockIdx.x * blockDim.x + threadIdx.x) >> 5;
    const int lane = threadIdx.x & 31;
    const int b     = wid >> 7;        // 128 M-tiles per batch (2048/16)
    const int mtile = wid & 127;

    const float* Qb = Q + (size_t)b * N_IN_PC * 3;
    const v2f*   Tp = (const v2f*)(Tpack + (size_t)b * nt * 4);

    const int r = lane & 15;
    const int h = lane >> 4;
    const int row = mtile * 16 + r;

    const float qx = Qb[row * 3 + 0];
    const float qy = Qb[row * 3 + 1];
    const float qz = Qb[row * 3 + 2];
    const float rnorm = qx * qx + qy * qy + qz * qz;

    v2f A;
    A[0] = h ? qz   : qx;
    A[1] = h ? 1.0f : qy;

    float rmin[8];
#pragma unroll
    for (int v = 0; v < 8; ++v) rmin[v] = 3.4028234663852886e38f;

    const int ntiles = nt >> 4;
    for (int tIdx = 0; tIdx < ntiles; ++tIdx) {
        const int col = (tIdx << 4) + r;
        const v2f Bm = Tp[(size_t)col * 2 + h];          // global_load_b64
        if (tIdx + 1 < ntiles)
            __builtin_prefetch((const float*)Tp + (size_t)(tIdx + 1) * 64, 0, 1);

        v8f cz = {0.f, 0.f, 0.f, 0.f, 0.f, 0.f, 0.f, 0.f};
        v8f d = __builtin_amdgcn_wmma_f32_16x16x4_f32(
            /*neg_a=*/false, A, /*neg_b=*/false, Bm,
            /*c_mod=*/(short)0, cz, /*reuse_a=*/false, /*reuse_b=*/false);
#pragma unroll
        for (int v = 0; v < 8; ++v) rmin[v] = fminf(rmin[v], d[v]);
    }

    // reduce min over the 16 lanes of each half, add |a|^2, sqrt, sum rows
    float s = 0.0f;
#pragma unroll
    for (int v = 0; v < 8; ++v) {
        float m = rmin[v];
        m = fminf(m, __shfl_xor(m, 1, 32));
        m = fminf(m, __shfl_xor(m, 2, 32));
        m = fminf(m, __shfl_xor(m, 4, 32));
        m = fminf(m, __shfl_xor(m, 8, 32));
        const float rn = __shfl(rnorm, v + 8 * h, 32);  // norm of row v+8h
        const float dd = fmaxf(m + rn, 0.0f);
        s += sqrtf(dd);
    }
    s += __shfl_xor(s, 16, 32);   // rows 0-7 half + rows 8-15 half
    if (lane == 0) part[wid] = s;
}

// ---------------------------------------------------------------------------
// Render losses: per-block partial sums (deterministic)
// ---------------------------------------------------------------------------
__global__ __launch_bounds__(256) void render_kernel(const float* __restrict__ cRaw,
                                                     const float* __restrict__ fRaw,
                                                     const float* __restrict__ tgt,
                                                     float* __restrict__ part) {
    const int N = BATCH * IMGSZ * IMGSZ;
    float sc = 0.0f, sf = 0.0f;
    for (int i = blockIdx.x * blockDim.x + threadIdx.x; i < N;
         i += gridDim.x * blockDim.x) {
        const float tg = tgt[i];
        const float vc = cRaw[i] * (1.0f / 255.0f);
        const float mk = (vc > 0.0f) ? 1.0f : 0.0f;
        const float dc = vc - tg;
        sc += dc * dc * mk;
        const float vf = fRaw[i] * (1.0f / 255.0f);
        const float df = vf - tg;
        sf += df * df;
    }
    __shared__ float rc[256], rf[256];
    rc[threadIdx.x] = sc;
    rf[threadIdx.x] = sf;
    __syncthreads();
    for (int off = 128; off > 0; off >>= 1) {
        if (threadIdx.x < off) {
            rc[threadIdx.x] += rc[threadIdx.x + off];
            rf[threadIdx.x] += rf[threadIdx.x + off];
        }
        __syncthreads();
    }
    if (threadIdx.x == 0) {
        part[blockIdx.x * 2 + 0] = rc[0];
        part[blockIdx.x * 2 + 1] = rf[0];
    }
}

// ---------------------------------------------------------------------------
// Final deterministic combine
// ws layout (floats): [0]denseSum8 [8]denseSsq8 [16]fpsSum8 [24]fpsSsq8
//                     [32]cdC 1024  [1056]cdF 1024  [2080]render 512
//                     [2592]samp 49152  [51744]packC 32768  [84512]packF 524288
// ---------------------------------------------------------------------------
__global__ void final_kernel(const float* __restrict__ w,
                             const float* __restrict__ dg,
                             const float* __restrict__ dr,
                             float* __restrict__ out) {
    if (threadIdx.x != 0 || blockIdx.x != 0) return;

    float dense_loss = 0.0f;
    for (int b = 0; b < BATCH; ++b) {
        const float s = w[b], q = w[8 + b];
        const float var = (q - s * s / (float)N_FINE) / (float)(N_FINE - 1);
        dense_loss += sqrtf(fmaxf(var, 0.0f));
    }
    dense_loss *= (1.0f / BATCH);

    float dense_loss_f = 0.0f;
    for (int b = 0; b < BATCH; ++b) {
        const float s = w[16 + b], q = w[24 + b];
        const float var = (q - s * s / (float)FPS_NS) / (float)(FPS_NS - 1);
        dense_loss_f += sqrtf(fmaxf(var, 0.0f));
    }
    dense_loss_f *= (1.0f / BATCH);

    float cdc = 0.0f, cdf = 0.0f;
    for (int i = 0; i < 1024; ++i) cdc += w[32 + i];
    for (int i = 0; i < 1024; ++i) cdf += w[1056 + i];
    const float inv = 1.0f / (float)(BATCH * N_IN_PC);
    const float g_cd = cdc * inv + cdf * inv;

    float rc = 0.0f, rf = 0.0f;
    for (int i = 0; i < 256; ++i) {
        rc += w[2080 + 2 * i + 0];
        rf += w[2080 + 2 * i + 1];
    }
    const float invR = 1.0f / (float)(BATCH * IMGSZ * IMGSZ);
    const float g_render = rc * invR + rf * invR;

    float ga = 0.0f, rl = 0.0f, fl = 0.0f;
    for (int b = 0; b < BATCH; ++b) {
        const float g = dg[b], rrv = dr[b];
        ga += (g - 1.0f) * (g - 1.0f);
        rl += (rrv - 1.0f) * (rrv - 1.0f);
        fl += g * g;
    }
    ga *= (1.0f / BATCH);
    rl *= (1.0f / BATCH);
    fl *= (1.0f / BATCH);

    out[0] = g_cd + g_render + (dense_loss + dense_loss_f) + ga;
    out[1] = 0.5f * (rl + fl);
}

// ---------------------------------------------------------------------------
extern "C" void kernel_launch(void* const* d_in, const int* in_sizes, int n_in,
                              void* d_out, int out_size, void* d_ws, size_t ws_size,
                              hipStream_t stream) {
    (void)in_sizes; (void)n_in; (void)out_size; (void)ws_size;
    const float* p_c    = (const float*)d_in[0];
    const float* p_f    = (const float*)d_in[1];
    const float* smap_c = (const float*)d_in[2];
    const float* smap_f = (const float*)d_in[3];
    const float* dense  = (const float*)d_in[6];
    const float* dg     = (const float*)d_in[8];
    const float* dr     = (const float*)d_in[9];
    const float* in_map = (const float*)d_in[10];
    const float* in_pc  = (const float*)d_in[11];
    float* w     = (float*)d_ws;
    float* samp  = w + 2592;
    float* packC = w + 51744;
    float* packF = w + 84512;

    fps_kernel<<<BATCH, 1024, 0, stream>>>(p_f, samp);
    density_kernel<<<BATCH, 1024, 0, stream>>>(samp, w + 16, w + 24);
    dense_kernel<<<BATCH, 1024, 0, stream>>>(dense, w + 0, w + 8);
    pack_kernel<<<(BATCH * N_COARSE + 255) / 256, 256, 0, stream>>>(p_c, BATCH * N_COARSE, packC);
    pack_kernel<<<(BATCH * N_FINE + 255) / 256, 256, 0, stream>>>(p_f, BATCH * N_FINE, packF);
    chamfer_kernel<<<128, 256, 0, stream>>>(in_pc, packC, N_COARSE, w + 32);
    chamfer_kernel<<<128, 256, 0, stream>>>(in_pc, packF, N_FINE, w + 1056);
    render_kernel<<<256, 256, 0, stream>>>(smap_c, smap_f, in_map, w + 2080);
    final_kernel<<<1, 1, 0, stream>>>(w, dg, dr, (float*)d_out);
}